// DGCNNBackbone_42133629173833
// MI455X (gfx1250) — compile-verified
//
#include <hip/hip_runtime.h>

typedef __attribute__((ext_vector_type(2))) float v2f;
typedef __attribute__((ext_vector_type(8))) float v8f;

#define EPS_BN   1e-5f
#define NSLOPE   0.2f
#define NEG_BIG  -3.0e38f
#define POS_BIG  3.0e38f

// D = A(16x4,f32) * B(4x16,f32) + C(16x16,f32), wave32.
static __device__ __forceinline__ v8f wmma_f32x4(v2f a, v2f b, v8f c) {
  return __builtin_amdgcn_wmma_f32_16x16x4_f32(
      false, a, false, b, (short)0, c, false, false);
}

// Sorted top-10 insertion, fully unrolled (compile-time indices -> registers).
static __device__ __forceinline__ void topk_insert(float (&vals)[10], int (&inds)[10],
                                                   float v, int m) {
  if (v > vals[9]) {
#pragma unroll
    for (int p = 9; p >= 0; --p) {
      float op  = vals[p];
      float opm = (p > 0) ? vals[p - 1] : POS_BIG;
      if (op < v) {
        if (opm >= v) { vals[p] = v; inds[p] = m; }
        else          { vals[p] = vals[p - 1]; inds[p] = inds[p - 1]; }
      }
    }
  }
}

// Load 2 consecutive floats; vectorized when the row stride keeps offsets even.
template <int CIN>
static __device__ __forceinline__ v2f load2(const float* __restrict__ p, size_t off) {
  if constexpr ((CIN & 1) == 0) {
    return *(const v2f*)(p + off);
  } else {
    v2f r; r.x = p[off]; r.y = p[off + 1]; return r;
  }
}

// ---------------------------------------------------------------------------
// Per-point squared norms: xx[b*N+n] = sum_c x[b,n,c]^2
// ---------------------------------------------------------------------------
__global__ void norms_kernel(const float* __restrict__ x, float* __restrict__ xx,
                             int cin, int total) {
  int i = blockIdx.x * blockDim.x + threadIdx.x;
  if (i >= total) return;
  const float* p = x + (size_t)i * cin;
  float s = 0.f;
  for (int c = 0; c < cin; ++c) s += p[c] * p[c];
  xx[i] = s;
}

// ---------------------------------------------------------------------------
// kNN (k=10) via WMMA fp32 Gram tiles.
// One wave per (batch, 16-row tile). A fragments register-cached; guard-free
// main K loop; lanes 0-15 scan tile cols 0-7, lanes 16-31 cols 8-15 with
// private top-10 lists merged at the end.
// ---------------------------------------------------------------------------
template <int CIN>
__global__ __launch_bounds__(32) void knn_kernel(const float* __restrict__ x,
                                                 const float* __restrict__ xx,
                                                 int* __restrict__ knnout,
                                                 int N) {
  static_assert(CIN % 4 == 0 || CIN % 4 == 1, "tail handling assumes rem 0 or 1");
  constexpr int  NSf  = CIN / 4;
  constexpr bool TAIL = (CIN % 4) != 0;
  const int lane  = threadIdx.x;
  const int b     = blockIdx.y;
  const int n0    = blockIdx.x * 16;
  const int l16   = lane & 15;
  const int kofs  = (lane < 16) ? 0 : 2;
  const int rhalf = (lane < 16) ? 0 : 8;
  const int cbase = (lane < 16) ? 0 : 8;
  const float* xb  = x  + (size_t)b * N * CIN;
  const float* xxb = xx + (size_t)b * N;

  // A fragments (rows n0..n0+15, all K) cached in registers — no guards in
  // the full steps: 4*s + kofs + 1 <= CIN-1 for s < CIN/4.
  v2f af[NSf + (TAIL ? 1 : 0)];
  {
    const size_t rb = (size_t)(n0 + l16) * CIN;
#pragma unroll
    for (int s = 0; s < NSf; ++s) af[s] = load2<CIN>(xb, rb + 4 * s + kofs);
    if constexpr (TAIL) {  // remainder 1: only column CIN-1, kofs==0 half
      v2f t; t.x = (kofs == 0) ? xb[rb + NSf * 4] : 0.f; t.y = 0.f;
      af[NSf] = t;
    }
  }
  float xxr[8];
#pragma unroll
  for (int r = 0; r < 8; ++r) xxr[r] = xxb[n0 + r + rhalf];

  float vals[10];
  int   inds[10];
#pragma unroll
  for (int i = 0; i < 10; ++i) { vals[i] = NEG_BIG; inds[i] = 0; }

  __shared__ float tile[16 * 17];
  __shared__ float mvs[16][10];
  __shared__ int   mis[16][10];

  for (int m0 = 0; m0 < N; m0 += 16) {
    v8f acc = 0.f;
    const int    col = m0 + l16;
    const size_t cb  = (size_t)col * CIN;
#pragma unroll
    for (int s = 0; s < NSf; ++s) {
      v2f bf = load2<CIN>(xb, cb + 4 * s + kofs);
      acc = wmma_f32x4(af[s], bf, acc);
    }
    if constexpr (TAIL) {
      v2f bf; bf.x = (kofs == 0) ? xb[cb + NSf * 4] : 0.f; bf.y = 0.f;
      acc = wmma_f32x4(af[NSf], bf, acc);
    }
    const float xxc = xxb[col];
#pragma unroll
    for (int r = 0; r < 8; ++r)
      tile[(r + rhalf) * 17 + l16] = 2.f * acc[r] - xxr[r] - xxc;
    __syncthreads();

    // All 32 lanes: lane handles row l16, 8 columns of its half.
#pragma unroll
    for (int c = 0; c < 8; ++c) {
      float v = tile[l16 * 17 + cbase + c];
      topk_insert(vals, inds, v, m0 + cbase + c);
    }
    __syncthreads();
  }

  // Merge upper-half lists into lower-half lists.
  if (lane >= 16) {
#pragma unroll
    for (int i = 0; i < 10; ++i) { mvs[l16][i] = vals[i]; mis[l16][i] = inds[i]; }
  }
  __syncthreads();
  if (lane < 16) {
#pragma unroll
    for (int i = 0; i < 10; ++i) topk_insert(vals, inds, mvs[l16][i], mis[l16][i]);
#pragma unroll
    for (int i = 0; i < 10; ++i)
      knnout[((size_t)b * N + n0 + lane) * 10 + i] = inds[i];
  }
}

// ---------------------------------------------------------------------------
// EdgeConv: per neighbor j, stage e=[x[m]-x[n], x[n]] (16 x K2, zero-padded)
// in LDS, WMMA against W^T tiles (guard-free main loop), BN + LeakyReLU,
// running max over j.  Block = (b, 16-point tile); wave w owns 16 channels.
// ---------------------------------------------------------------------------
template <int CIN>
__global__ void edgeconv_kernel(const float* __restrict__ x, const int* __restrict__ knn,
                                const float* __restrict__ w, const float* __restrict__ g,
                                const float* __restrict__ bta, const float* __restrict__ rm,
                                const float* __restrict__ rv, float* __restrict__ out,
                                int cout, int N) {
  constexpr int  K2   = 2 * CIN;
  constexpr int  NSf  = K2 / 4;
  constexpr bool TAIL = (K2 % 4) != 0;
  constexpr int  K2r  = 4 * NSf + (TAIL ? 4 : 0);  // LDS columns (zero padded)
  constexpr int  est  = K2 + 4;                    // row stride, conflict-free
  extern __shared__ float sm[];
  const int b     = blockIdx.y;
  const int n0    = blockIdx.x * 16;
  const int lane  = threadIdx.x & 31;
  const int wave  = threadIdx.x >> 5;
  const int l16   = lane & 15;
  const int kofs  = (lane < 16) ? 0 : 2;
  const int rhalf = (lane < 16) ? 0 : 8;
  const int oc    = wave * 16 + l16;
  const float* xb = x + (size_t)b * N * CIN;
  const float* wr = w + (size_t)oc * K2;

  const float sc = g[oc] * rsqrtf(rv[oc] + EPS_BN);
  const float sh = bta[oc] - rm[oc] * sc;

  float maxv[8];
#pragma unroll
  for (int r = 0; r < 8; ++r) maxv[r] = NEG_BIG;

  for (int j = 0; j < 10; ++j) {
    // Build e-tile (16 x K2r) in LDS cooperatively; cols [K2,K2r) zeroed.
    for (int t = threadIdx.x; t < 16 * K2r; t += blockDim.x) {
      int r = t / K2r, c = t - r * K2r;
      int n = n0 + r;
      float val;
      if (c < CIN) {
        int m = knn[((size_t)b * N + n) * 10 + j];
        val = xb[(size_t)m * CIN + c] - xb[(size_t)n * CIN + c];
      } else if (c < K2) {
        val = xb[(size_t)n * CIN + (c - CIN)];
      } else {
        val = 0.f;
      }
      sm[r * est + c] = val;
    }
    __syncthreads();

    v8f acc = 0.f;
    const float* arow = sm + l16 * est;
#pragma unroll 8
    for (int s = 0; s < NSf; ++s) {
      int k0 = 4 * s + kofs;
      v2f a  = *(const v2f*)(arow + k0);   // est even, k0 even -> aligned
      v2f bb = *(const v2f*)(wr + k0);     // K2 even, k0 even -> aligned
      acc = wmma_f32x4(a, bb, acc);
    }
    if constexpr (TAIL) {  // K2 % 4 == 2: last two K columns on kofs==0 half
      int k0 = 4 * NSf + kofs;
      v2f a = *(const v2f*)(arow + k0);    // zero-padded region for kofs==2
      v2f bb;
      if (kofs == 0) bb = *(const v2f*)(wr + k0);
      else { bb.x = 0.f; bb.y = 0.f; }
      acc = wmma_f32x4(a, bb, acc);
    }
#pragma unroll
    for (int r = 0; r < 8; ++r) {
      float h = acc[r] * sc + sh;
      h = (h >= 0.f) ? h : NSLOPE * h;
      maxv[r] = fmaxf(maxv[r], h);
    }
    __syncthreads();
  }

#pragma unroll
  for (int r = 0; r < 8; ++r)
    out[((size_t)b * N + n0 + r + rhalf) * cout + oc] = maxv[r];
}

// ---------------------------------------------------------------------------
// Global max over N of channel-concatenated [x1|x2|x3|x4] -> gfeat (16 x 512),
// rows 8..15 zeroed so the linear kernel's A fragments need no guards.
// ---------------------------------------------------------------------------
__global__ void gmax_kernel(const float* __restrict__ x1, const float* __restrict__ x2,
                            const float* __restrict__ x3, const float* __restrict__ x4,
                            float* __restrict__ gfeat, int N) {
  int b = blockIdx.x;  // 0..15
  int c = threadIdx.x; // 0..511
  if (b >= 8) { gfeat[b * 512 + c] = 0.f; return; }
  const float* src; int co, cl;
  if      (c < 64)  { src = x1; co = 64;  cl = c; }
  else if (c < 128) { src = x2; co = 64;  cl = c - 64; }
  else if (c < 256) { src = x3; co = 128; cl = c - 128; }
  else              { src = x4; co = 256; cl = c - 256; }
  const float* p = src + (size_t)b * N * co + cl;
  float m0 = NEG_BIG, m1 = NEG_BIG, m2 = NEG_BIG, m3 = NEG_BIG;
  for (int n = 0; n < N; n += 4) {  // N multiple of 4; 4-way MLP
    m0 = fmaxf(m0, p[(size_t)(n + 0) * co]);
    m1 = fmaxf(m1, p[(size_t)(n + 1) * co]);
    m2 = fmaxf(m2, p[(size_t)(n + 2) * co]);
    m3 = fmaxf(m3, p[(size_t)(n + 3) * co]);
  }
  gfeat[b * 512 + c] = fmaxf(fmaxf(m0, m1), fmaxf(m2, m3));
}

// ---------------------------------------------------------------------------
// Final linear 512->1024 + bias + BN + LeakyReLU via WMMA.
// M dimension = batch; gfeat is zero-padded to 16 rows -> guard-free loads.
// ---------------------------------------------------------------------------
__global__ __launch_bounds__(32) void linear_kernel(
    const float* __restrict__ gfeat, const float* __restrict__ lw,
    const float* __restrict__ lb, const float* __restrict__ g,
    const float* __restrict__ bt, const float* __restrict__ rm,
    const float* __restrict__ rv, float* __restrict__ out) {
  const int lane = threadIdx.x;
  const int l16  = lane & 15;
  const int kofs = (lane < 16) ? 0 : 2;
  const int oc   = blockIdx.x * 16 + l16;
  const float* ar = gfeat + l16 * 512;
  const float* br = lw + (size_t)oc * 512;

  v8f acc = 0.f;
#pragma unroll 8
  for (int s = 0; s < 128; ++s) {
    int k0 = 4 * s + kofs;
    v2f a  = *(const v2f*)(ar + k0);
    v2f bb = *(const v2f*)(br + k0);
    acc = wmma_f32x4(a, bb, acc);
  }

  const float sc = g[oc] * rsqrtf(rv[oc] + EPS_BN);
  const float sh = bt[oc] - rm[oc] * sc + lb[oc] * sc;

  if (lane < 16) {  // lanes 0..15 hold M = r = batch index (8 valid rows)
#pragma unroll
    for (int r = 0; r < 8; ++r) {
      float h = acc[r] * sc + sh;
      h = (h >= 0.f) ? h : NSLOPE * h;
      out[r * 1024 + oc] = h;
    }
  }
}

// ---------------------------------------------------------------------------
// x4 (B,N,256) -> out (B,256,N), LDS-tiled for coalescing both ways.
// ---------------------------------------------------------------------------
__global__ void transpose_kernel(const float* __restrict__ x4, float* __restrict__ out,
                                 int N, int C) {
  __shared__ float t[16][17];
  int b = blockIdx.z;
  int n0 = blockIdx.x * 16, c0 = blockIdx.y * 16;
  int tx = threadIdx.x, ty = threadIdx.y;
  t[ty][tx] = x4[((size_t)b * N + n0 + ty) * C + c0 + tx];
  __syncthreads();
  out[((size_t)b * C + c0 + ty) * N + n0 + tx] = t[tx][ty];
}

// ---------------------------------------------------------------------------
extern "C" void kernel_launch(void* const* d_in, const int* in_sizes, int n_in,
                              void* d_out, int out_size, void* d_ws, size_t ws_size,
                              hipStream_t stream) {
  (void)in_sizes; (void)n_in; (void)out_size; (void)ws_size;
  const int B = 8, N = 2048;

  const float* x = (const float*)d_in[0];
  const float* w[4]  = {(const float*)d_in[1],  (const float*)d_in[6],
                        (const float*)d_in[11], (const float*)d_in[16]};
  const float* gm[4] = {(const float*)d_in[2],  (const float*)d_in[7],
                        (const float*)d_in[12], (const float*)d_in[17]};
  const float* bt[4] = {(const float*)d_in[3],  (const float*)d_in[8],
                        (const float*)d_in[13], (const float*)d_in[18]};
  const float* rm[4] = {(const float*)d_in[4],  (const float*)d_in[9],
                        (const float*)d_in[14], (const float*)d_in[19]};
  const float* rv[4] = {(const float*)d_in[5],  (const float*)d_in[10],
                        (const float*)d_in[15], (const float*)d_in[20]};
  const float* lw  = (const float*)d_in[21];
  const float* lb  = (const float*)d_in[22];
  const float* g5  = (const float*)d_in[23];
  const float* b5  = (const float*)d_in[24];
  const float* rm5 = (const float*)d_in[25];
  const float* rv5 = (const float*)d_in[26];

  // Workspace layout (bytes)
  char* ws = (char*)d_ws;
  int*   idx   = (int*)ws;                                        // 655360
  float* xx    = (float*)(ws + 655360);                           // 65536
  float* x1    = (float*)(ws + 720896);                           // 4 MiB
  float* x2    = (float*)(ws + 720896 + 4194304);
  float* x3    = (float*)(ws + 720896 + 2 * 4194304);
  float* x4    = (float*)(ws + 720896 + 2 * 4194304 + 8388608);
  float* gfeat = (float*)(ws + 720896 + 2 * 4194304 + 8388608 + 16777216); // 16x512

  const int cins[4]  = {5, 64, 64, 128};
  const int couts[4] = {64, 64, 128, 256};
  float* xouts[4] = {x1, x2, x3, x4};
  const float* xin = x;

  for (int L = 0; L < 4; ++L) {
    int ci = cins[L], co = couts[L];
    norms_kernel<<<dim3((B * N + 255) / 256), 256, 0, stream>>>(xin, xx, ci, B * N);
    dim3 kg(N / 16, B);
    int threads  = (co / 16) * 32;
    size_t shmem = (size_t)16 * (2 * ci + 4) * sizeof(float);
    switch (L) {
      case 0:
        knn_kernel<5><<<kg, 32, 0, stream>>>(xin, xx, idx, N);
        edgeconv_kernel<5><<<kg, threads, shmem, stream>>>(
            xin, idx, w[L], gm[L], bt[L], rm[L], rv[L], xouts[L], co, N);
        break;
      case 1:
        knn_kernel<64><<<kg, 32, 0, stream>>>(xin, xx, idx, N);
        edgeconv_kernel<64><<<kg, threads, shmem, stream>>>(
            xin, idx, w[L], gm[L], bt[L], rm[L], rv[L], xouts[L], co, N);
        break;
      case 2:
        knn_kernel<64><<<kg, 32, 0, stream>>>(xin, xx, idx, N);
        edgeconv_kernel<64><<<kg, threads, shmem, stream>>>(
            xin, idx, w[L], gm[L], bt[L], rm[L], rv[L], xouts[L], co, N);
        break;
      case 3:
        knn_kernel<128><<<kg, 32, 0, stream>>>(xin, xx, idx, N);
        edgeconv_kernel<128><<<kg, threads, shmem, stream>>>(
            xin, idx, w[L], gm[L], bt[L], rm[L], rv[L], xouts[L], co, N);
        break;
    }
    xin = xouts[L];
  }

  gmax_kernel<<<16, 512, 0, stream>>>(x1, x2, x3, x4, gfeat, N);
  linear_kernel<<<1024 / 16, 32, 0, stream>>>(gfeat, lw, lb, g5, b5, rm5, rv5,
                                              (float*)d_out);
  transpose_kernel<<<dim3(N / 16, 256 / 16, B), dim3(16, 16), 0, stream>>>(
      x4, ((float*)d_out) + 8192, N, 256);
}